// Network_77017353551897
// MI455X (gfx1250) — compile-verified
//
#include <hip/hip_runtime.h>

typedef __attribute__((ext_vector_type(2))) float v2f;
typedef __attribute__((ext_vector_type(8))) float v8f;

#define EMB 768
#define MSEG 256
#define NMEM 512
#define NB 32
#define NC 8
#define NJ 9            // 1 question column + 8 answer-choice columns
#define TBL_STRIDE (50000L * 256L)

// ---------------------------------------------------------------------------
// fp32 WMMA tile GEMM:  D[m][n] = sum_k X[m][k] * Wm[n][k]   (D = X @ Wm^T)
// grid = (M/16, EMB/16), block = 32 (one wave per 16x16 output tile)
// ---------------------------------------------------------------------------
__global__ void gemm_xwt_kernel(const float* __restrict__ X,
                                const float* __restrict__ Wm,
                                float* __restrict__ D) {
    int rt = blockIdx.x, nt = blockIdx.y;
    int lane = threadIdx.x;
    int r = lane & 15, half = lane >> 4;      // A: M=r, K=k0+2*half ; B: N=r
    const float* xrow = X  + (rt * 16 + r) * EMB + half * 2;
    const float* wrow = Wm + (nt * 16 + r) * EMB + half * 2;
    v8f acc = {};
#pragma unroll 4
    for (int k0 = 0; k0 < EMB; k0 += 4) {
        v2f af = *(const v2f*)(xrow + k0);
        v2f bf = *(const v2f*)(wrow + k0);
        acc = __builtin_amdgcn_wmma_f32_16x16x4_f32(false, af, false, bf,
                                                    (short)0, acc, false, false);
    }
    int n = nt * 16 + r;
    int mbase = rt * 16 + half * 8;
#pragma unroll
    for (int v = 0; v < 8; ++v)
        D[(mbase + v) * EMB + n] = acc[v];
}

// ---------------------------------------------------------------------------
// Fused embed-gather + WMMA scores:
//  scores[b][m][j] = sum_d ( sum_w A[idx_seg(b,m,w)][k] )_{d=seg*256+k} * T[j][b][d]
// grid = (32 b, 32 m-tiles), block = 128 (4 waves). LDS tile 16x768 fp32.
// ---------------------------------------------------------------------------
__global__ void scores_kernel(const float* __restrict__ table,
                              const int* __restrict__ subs,
                              const int* __restrict__ rels,
                              const int* __restrict__ objs,
                              const float* __restrict__ T,
                              float* __restrict__ scores) {
    int b = blockIdx.x, mt = blockIdx.y;
    __shared__ float tile[16][EMB];     // 48 KB
    __shared__ float part[4][16][16];   // 4 KB partial D tiles
    __shared__ int   sidx[3][16][8];    // 1.5 KB indices
    int t = threadIdx.x;

    // phase 0: stage indices for these 16 memories
    for (int i = t; i < 3 * 16 * 8; i += 128) {
        int seg = i / 128, rem = i % 128, ml = rem / 8, w = rem % 8;
        const int* ia = (seg == 0) ? subs : (seg == 1) ? rels : objs;
        sidx[seg][ml][w] = ia[b * (NMEM * 8) + (mt * 16 + ml) * 8 + w];
    }
    __syncthreads();

    // phase 1: gather-sum 8 word rows per segment -> LDS tile [16][768]
    for (int i = t; i < 16 * EMB; i += 128) {
        int ml = i / EMB, dk = i % EMB;
        int seg = dk >> 8, k = dk & 255;
        float acc = 0.f;
#pragma unroll
        for (int w = 0; w < 8; ++w)
            acc += table[sidx[seg][ml][w] * MSEG + k];
        tile[ml][dk] = acc;
    }
    __syncthreads();

    // phase 2: WMMA (A = LDS tile 16xK, B = T columns 16 j), K split over 4 waves
    int wave = t >> 5, lane = t & 31, r = lane & 15, half = lane >> 4;
    v8f acc = {};
    const float* tb = T + r * (NB * EMB) + b * EMB + half * 2;  // B[k][j=r] = T[j][b][k]
    int kb = wave * 192;
    for (int s = 0; s < 48; ++s) {
        int k0 = kb + s * 4;
        v2f af = *(const v2f*)&tile[r][k0 + half * 2];
        v2f bf = *(const v2f*)(tb + k0);
        acc = __builtin_amdgcn_wmma_f32_16x16x4_f32(false, af, false, bf,
                                                    (short)0, acc, false, false);
    }
#pragma unroll
    for (int v = 0; v < 8; ++v)
        part[wave][v + half * 8][r] = acc[v];
    __syncthreads();

    // phase 3: reduce 4 partials, emit scores[b][m][j] (j>=9 columns are zero)
    for (int i = t; i < 256; i += 128) {
        int mp = i >> 4, n = i & 15;
        float s = part[0][mp][n] + part[1][mp][n] + part[2][mp][n] + part[3][mp][n];
        scores[(b * NMEM + mt * 16 + mp) * 16 + n] = s;
    }
}

// ---------------------------------------------------------------------------
__device__ __forceinline__ float block_red(float v, float* red, int mode) {
    int t = threadIdx.x;
    red[t] = v;
    __syncthreads();
    for (int s = 64; s > 0; s >>= 1) {
        if (t < s) {
            float a = red[t], b = red[t + s];
            red[t] = (mode == 0) ? fmaxf(a, b) : (mode == 2) ? fminf(a, b) : (a + b);
        }
        __syncthreads();
    }
    float r = red[0];
    __syncthreads();
    return r;
}

// renorm over M=512 in-place; j==0 uses the question variant, j>=1 the answer one.
__global__ void renorm_kernel(float* __restrict__ scores) {
    int b = blockIdx.x, j = blockIdx.y;  // j in 0..8
    __shared__ float red[128];
    int t = threadIdx.x;
    float s[4];
#pragma unroll
    for (int i = 0; i < 4; ++i)
        s[i] = scores[(b * NMEM + (t + 128 * i)) * 16 + j];

    float mx = fmaxf(fmaxf(s[0], s[1]), fmaxf(s[2], s[3]));
    mx = block_red(mx, red, 0);
    float se = 0.f;
#pragma unroll
    for (int i = 0; i < 4; ++i) se += expf(s[i] - mx);
    se = block_red(se, red, 1);
    float lse = mx + logf(se);
    float p[4];
#pragma unroll
    for (int i = 0; i < 4; ++i) p[i] = s[i] - lse;

    if (j == 0) {  // question: shift by row-min, L1 renorm with 1e-8 offset
        float mn = fminf(fminf(p[0], p[1]), fminf(p[2], p[3]));
        mn = block_red(mn, red, 2);
#pragma unroll
        for (int i = 0; i < 4; ++i) p[i] -= mn;
        float nm = 0.f;
#pragma unroll
        for (int i = 0; i < 4; ++i) nm += fabsf(p[i] + 1e-8f);
        nm = block_red(nm, red, 1);
#pragma unroll
        for (int i = 0; i < 4; ++i) p[i] /= nm;
    } else {       // answers: L1 renorm with zero-norm guard
        float nm = 0.f;
#pragma unroll
        for (int i = 0; i < 4; ++i) nm += fabsf(p[i]);
        nm = block_red(nm, red, 1);
        if (nm == 0.f) {  // uniform branch
#pragma unroll
            for (int i = 0; i < 4; ++i) p[i] += 1.f;
            float nm2 = 0.f;
#pragma unroll
            for (int i = 0; i < 4; ++i) nm2 += fabsf(p[i]);
            nm = block_red(nm2, red, 1);
        }
#pragma unroll
        for (int i = 0; i < 4; ++i) p[i] /= nm;
    }
#pragma unroll
    for (int i = 0; i < 4; ++i)
        scores[(b * NMEM + (t + 128 * i)) * 16 + j] = p[i];
}

// ---------------------------------------------------------------------------
// Fused gather for o:  o_all[j][b][seg*256+k] = sum_{m,w} P[b][m][j] * A2[idx][k]
// grid = (32 b, 3 seg), block = 256 (k = tid)
// ---------------------------------------------------------------------------
__global__ void ogather_kernel(const float* __restrict__ table,
                               const int* __restrict__ subs,
                               const int* __restrict__ rels,
                               const int* __restrict__ objs,
                               const float* __restrict__ P,
                               float* __restrict__ o_all) {
    int b = blockIdx.x, seg = blockIdx.y;
    const int* ia = (seg == 0) ? subs : (seg == 1) ? rels : objs;
    __shared__ float Pl[64][NJ];
    __shared__ int   idxl[64][8];
    int t = threadIdx.x, k = t;
    float acc[NJ];
#pragma unroll
    for (int j = 0; j < NJ; ++j) acc[j] = 0.f;

    for (int c0 = 0; c0 < NMEM; c0 += 64) {
        for (int i = t; i < 64 * NJ; i += 256) {
            int mm = i / NJ, j = i % NJ;
            Pl[mm][j] = P[(b * NMEM + c0 + mm) * 16 + j];
        }
        for (int i = t; i < 64 * 8; i += 256) {
            int mm = i / 8, w = i % 8;
            idxl[mm][w] = ia[b * (NMEM * 8) + (c0 + mm) * 8 + w];
        }
        __syncthreads();
        for (int mm = 0; mm < 64; ++mm) {
#pragma unroll
            for (int w = 0; w < 8; ++w) {
                float val = table[idxl[mm][w] * MSEG + k];
#pragma unroll
                for (int j = 0; j < NJ; ++j) acc[j] += Pl[mm][j] * val;
            }
        }
        __syncthreads();
    }
#pragma unroll
    for (int j = 0; j < NJ; ++j)
        o_all[(j * NB + b) * EMB + seg * MSEG + k] = acc[j];
}

// ---------------------------------------------------------------------------
__global__ void init_u_kernel(const float* __restrict__ Bt,
                              const int* __restrict__ ques, float* __restrict__ u) {
    int b = blockIdx.x;
    for (int rep = 0; rep < 3; ++rep) {
        int d = threadIdx.x + rep * 256;
        float acc = 0.f;
        for (int w = 0; w < 32; ++w) acc += Bt[ques[b * 32 + w] * EMB + d];
        u[b * EMB + d] = acc;
    }
}

__global__ void init_a_kernel(const float* __restrict__ Bt,
                              const int* __restrict__ ac, float* __restrict__ a) {
    int b = blockIdx.x, c = blockIdx.y;
    for (int rep = 0; rep < 3; ++rep) {
        int d = threadIdx.x + rep * 256;
        float acc = 0.f;
#pragma unroll
        for (int w = 0; w < 8; ++w) acc += Bt[ac[(b * NC + c) * 8 + w] * EMB + d];
        a[(c * NB + b) * EMB + d] = acc;   // layout [c][b][d]
    }
}

__global__ void zero_kernel(float* __restrict__ p, int n) {
    int i = blockIdx.x * 256 + threadIdx.x;
    if (i < n) p[i] = 0.f;
}

// u += o_all[j=0]; a += o_all[j=1..8]  (layouts match flat)
__global__ void update_kernel(const float* __restrict__ o_all,
                              float* __restrict__ u, float* __restrict__ a) {
    int i = blockIdx.x * 256 + threadIdx.x;
    const int N1 = NB * EMB;
    const int NT = N1 + NC * NB * EMB;
    if (i < N1) u[i] += o_all[i];
    else if (i < NT) a[i - N1] += o_all[i];
}

// pred[b][c] = o_q[b] . Wo[c*32+b]
__global__ void pred_kernel(const float* __restrict__ oq,
                            const float* __restrict__ Wo, float* __restrict__ out) {
    int m = blockIdx.x;  // m = c*32 + b
    int t = threadIdx.x; // 128
    __shared__ float red[128];
    float s = 0.f;
    for (int d = t; d < EMB; d += 128)
        s += oq[(m & 31) * EMB + d] * Wo[m * EMB + d];
    s = block_red(s, red, 1);
    if (t == 0) out[(m & 31) * NC + (m >> 5)] = s;
}

// ---------------------------------------------------------------------------
extern "C" void kernel_launch(void* const* d_in, const int* in_sizes, int n_in,
                              void* d_out, int out_size, void* d_ws, size_t ws_size,
                              hipStream_t stream) {
    (void)in_sizes; (void)n_in; (void)out_size; (void)ws_size;
    const int*   subs = (const int*)d_in[0];
    const int*   rels = (const int*)d_in[1];
    const int*   objs = (const int*)d_in[2];
    const int*   ques = (const int*)d_in[3];
    const int*   ach  = (const int*)d_in[4];
    const float* At   = (const float*)d_in[5];
    const float* Bt   = (const float*)d_in[6];
    const float* U    = (const float*)d_in[7];
    const float* V    = (const float*)d_in[8];
    const float* W    = (const float*)d_in[9];
    float* out = (float*)d_out;

    float* u      = (float*)d_ws;               // [32][768]
    float* a      = u + NB * EMB;               // [8][32][768]
    float* T      = a + NC * NB * EMB;          // [16][32][768] (j-major RHS, padded)
    float* scores = T + 16 * NB * EMB;          // [32][512][16]
    float* o_all  = scores + NB * NMEM * 16;    // [9][32][768]
    float* Wo     = o_all + NJ * NB * EMB;      // [8*32][768]

    init_u_kernel<<<NB, 256, 0, stream>>>(Bt, ques, u);
    init_a_kernel<<<dim3(NB, NC), 256, 0, stream>>>(Bt, ach, a);
    zero_kernel<<<(16 * NB * EMB + 255) / 256, 256, 0, stream>>>(T, 16 * NB * EMB);

    for (int hop = 0; hop < 2; ++hop) {
        const float* tin  = At + hop * TBL_STRIDE;
        const float* tout = At + (hop + 1) * TBL_STRIDE;
        // t_q = u @ U^T  -> T[0][b][:]
        gemm_xwt_kernel<<<dim3(NB / 16, EMB / 16), 32, 0, stream>>>(u, U, T);
        // t_a = a @ V^T  -> T[1+c][b][:]
        gemm_xwt_kernel<<<dim3(NC * NB / 16, EMB / 16), 32, 0, stream>>>(a, V, T + NB * EMB);
        // fused gather + WMMA scores
        scores_kernel<<<dim3(NB, NMEM / 16), 128, 0, stream>>>(tin, subs, rels, objs, T, scores);
        // renorm (q column + 8 answer columns), in place
        renorm_kernel<<<dim3(NB, NJ), 128, 0, stream>>>(scores);
        // fused gather o = P^T . out
        ogather_kernel<<<dim3(NB, 3), 256, 0, stream>>>(tout, subs, rels, objs, scores, o_all);
        // u += o_q ; a += o_a
        update_kernel<<<((NB + NC * NB) * EMB + 255) / 256, 256, 0, stream>>>(o_all, u, a);
    }

    // Wo = o_a @ W^T ; pred[b][c] = o_q[b] . Wo[c][b]
    gemm_xwt_kernel<<<dim3(NC * NB / 16, EMB / 16), 32, 0, stream>>>(o_all + NB * EMB, W, Wo);
    pred_kernel<<<NC * NB, 128, 0, stream>>>(o_all, Wo, out);
}